// PersLay_30270929502733
// MI455X (gfx1250) — compile-verified
//
#include <hip/hip_runtime.h>

// PersLay forward for MI455X (gfx1250, wave32).
//
// Stage 1+2 (dominant): 64M v_exp_f32 -> transcendental-bound; minimize
//   per-(p,m) overhead, exploit sorted segment_ids for cheap reduction.
// Stage 3: feats[2048,64] @ fc_w.T[64,10] + b via V_WMMA_F32_16X16X4_F32.

typedef float v2f __attribute__((ext_vector_type(2)));
typedef float v8f __attribute__((ext_vector_type(8)));

#define NSEG  2048
#define M_DIM 64
#define L_DIM 10

// -------------------------------------------------------------------------
// Kernel 0: zero the feats accumulator (d_ws is poisoned by the harness).
// -------------------------------------------------------------------------
__global__ void zero_kernel(float* __restrict__ p, int n) {
    int i = blockIdx.x * blockDim.x + threadIdx.x;
    if (i < n) p[i] = 0.0f;
}

// -------------------------------------------------------------------------
// Kernel 1: gaussian kernel + segment sum into feats[NSEG][M_DIM].
// Each wave handles 256 consecutive points (8 rounds x 32 lanes).
//   k = exp(-2*d2) = exp2( L2E*(-2pp - 2tt + 4 px*tx + 4 py*ty) )
// Constants pre-scaled by L2E so the inner loop is 2 fma + 1 v_exp_f32;
// per-m constants packed as float4 in LDS -> single ds_load_b128 per m.
// -------------------------------------------------------------------------
__global__ __launch_bounds__(256) void feats_kernel(
    const float* __restrict__ points, const int* __restrict__ segids,
    const float* __restrict__ theta, float* __restrict__ feats, int P) {

    __shared__ float4 s_t[M_DIM];          // (ax, ay, c, 0) per center
    const float L2E = 1.4426950408889634f;

    int tid = threadIdx.x;
    if (tid < M_DIM) {
        float tx = theta[2 * tid + 0], ty = theta[2 * tid + 1];
        s_t[tid] = make_float4(4.0f * L2E * tx,
                               4.0f * L2E * ty,
                               -2.0f * L2E * (tx * tx + ty * ty),
                               0.0f);
    }
    __syncthreads();

    const int lane = tid & 31;
    const int wave = tid >> 5;
    long base = ((long)blockIdx.x * 8 + wave) * 256;   // 256 points per wave

    float px[8], py[8], cp[8];
    int   seg[8];
#pragma unroll
    for (int i = 0; i < 8; ++i) {
        long p = base + (long)i * 32 + lane;
        if (p < (long)P) {
            float x = points[2 * p], y = points[2 * p + 1];
            px[i] = x; py[i] = y;
            cp[i] = -2.0f * L2E * (x * x + y * y);
            seg[i] = segids[p];
        } else {                       // dead lane: exp2(-1e30) == 0
            px[i] = 0.0f; py[i] = 0.0f; cp[i] = -1.0e30f; seg[i] = NSEG;
        }
    }

    // Chunk-wide uniformity test (min/max butterfly across the wave).
    int smin = seg[0], smax = seg[0];
#pragma unroll
    for (int i = 1; i < 8; ++i) { smin = min(smin, seg[i]); smax = max(smax, seg[i]); }
#pragma unroll
    for (int off = 16; off >= 1; off >>= 1) {
        smin = min(smin, __shfl_xor(smin, off));
        smax = max(smax, __shfl_xor(smax, off));
    }

    if (smin == smax && smax < NSEG) {
        // ---- fast path: 256 points all in one segment ----
        float* frow = feats + (long)smin * M_DIM;
#pragma unroll 4
        for (int m = 0; m < M_DIM; ++m) {
            float4 t = s_t[m];
            float acc = 0.0f;
#pragma unroll
            for (int i = 0; i < 8; ++i)
                acc += __builtin_amdgcn_exp2f(
                    fmaf(t.x, px[i], fmaf(t.y, py[i], t.z + cp[i])));
#pragma unroll
            for (int off = 16; off >= 1; off >>= 1)
                acc += __shfl_down(acc, off);
            if (lane == 0) atomicAdd(frow + m, acc);
        }
    } else {
        // ---- boundary path: per-round segmented shuffle scan ----
#pragma unroll 1
        for (int i = 0; i < 8; ++i) {
            int s = seg[i];
            bool pred[5];
#pragma unroll
            for (int b = 0; b < 5; ++b) {
                int off = 1 << b;
                int sUp = __shfl_up(s, off);
                pred[b] = (lane >= off) && (sUp == s);
            }
            int  sNext = __shfl_down(s, 1);
            bool tail  = (lane == 31) || (sNext != s);
            bool valid = (s < NSEG);
            float* frow = feats + (long)s * M_DIM;
            float axp = px[i], ayp = py[i], cpl = cp[i];
#pragma unroll 2
            for (int m = 0; m < M_DIM; ++m) {
                float4 t = s_t[m];
                float k = __builtin_amdgcn_exp2f(
                    fmaf(t.x, axp, fmaf(t.y, ayp, t.z + cpl)));
#pragma unroll
                for (int b = 0; b < 5; ++b) {
                    float u = __shfl_up(k, 1 << b);
                    if (pred[b]) k += u;
                }
                if (tail && valid) atomicAdd(frow + m, k);
            }
        }
    }
}

// -------------------------------------------------------------------------
// Kernel 2: out[2048,10] = feats[2048,64] @ fc_w.T + fc_b
// One wave per 16-row tile; 16x V_WMMA_F32_16X16X4_F32 over K=64.
// Layouts per CDNA5 ISA 7.12.2:
//   A 16x4 f32 : lane&15 = M row; half h: VGPR0=K(2h), VGPR1=K(2h+1)
//   B 4x16 f32 : lane&15 = N col; half h: VGPR0=row(2h), VGPR1=row(2h+1)
//   C 16x16 f32: VGPR j -> M = j + 8h, N = lane&15
// Padding columns (10..15) handled by clamped in-bounds loads + v_cndmask
// (no divergent branch diamonds around the loads).
// -------------------------------------------------------------------------
__global__ __launch_bounds__(32) void head_wmma_kernel(
    const float* __restrict__ feats, const float* __restrict__ fcw,
    const float* __restrict__ fcb, float* __restrict__ out) {

    const int lane = threadIdx.x & 31;
    const int col  = lane & 15;
    const int h    = lane >> 4;
    const long row0 = (long)blockIdx.x * 16;

    const bool bcol = (col < L_DIM);
    const int  ccol = bcol ? col : (L_DIM - 1);   // always in-bounds

    float bias = fcb[ccol];
    bias = bcol ? bias : 0.0f;
    v8f c;
#pragma unroll
    for (int j = 0; j < 8; ++j) c[j] = bias;

    const float* aptr = feats + (row0 + col) * M_DIM;   // A row = lane&15
    const float* bptr = fcw + (long)ccol * M_DIM;       // B[k][n] = fc_w[n][k]

#pragma unroll
    for (int kk = 0; kk < M_DIM; kk += 4) {
        int k0 = kk + 2 * h;                      // even -> 8B aligned
        v2f a = *(const v2f*)(aptr + k0);
        v2f b = *(const v2f*)(bptr + k0);
        b.x = bcol ? b.x : 0.0f;                  // v_cndmask, no branch
        b.y = bcol ? b.y : 0.0f;
        c = __builtin_amdgcn_wmma_f32_16x16x4_f32(
                /*neg_a=*/false, a, /*neg_b=*/false, b,
                /*c_mod=*/(short)0, c, /*reuse_a=*/false, /*reuse_b=*/false);
    }

#pragma unroll
    for (int j = 0; j < 8; ++j) {
        long row = row0 + j + 8 * h;
        if (bcol) out[row * L_DIM + col] = c[j];
    }
}

// -------------------------------------------------------------------------
extern "C" void kernel_launch(void* const* d_in, const int* in_sizes, int n_in,
                              void* d_out, int out_size, void* d_ws, size_t ws_size,
                              hipStream_t stream) {
    const float* points = (const float*)d_in[0];
    const int*   segids = (const int*)d_in[1];
    const float* theta  = (const float*)d_in[2];
    const float* fcw    = (const float*)d_in[3];
    const float* fcb    = (const float*)d_in[4];
    float* out   = (float*)d_out;
    float* feats = (float*)d_ws;           // NSEG * M_DIM floats = 512 KB

    const int P = in_sizes[0] / 2;

    const int nfeat = NSEG * M_DIM;
    zero_kernel<<<(nfeat + 255) / 256, 256, 0, stream>>>(feats, nfeat);

    const int ptsPerBlock = 2048;          // 8 waves x 256 points
    const int nb = (P + ptsPerBlock - 1) / ptsPerBlock;
    feats_kernel<<<nb, 256, 0, stream>>>(points, segids, theta, feats, P);

    head_wmma_kernel<<<NSEG / 16, 32, 0, stream>>>(feats, fcw, fcb, out);
}